// ETF_28200755266134
// MI455X (gfx1250) — compile-verified
//
#include <hip/hip_runtime.h>
#include <stdint.h>

// -------- problem constants (fixed by reference: (8,1,1024,1024) fp32) -----
#define MU    5
#define ETA   1.0f
#define ITERS 3
#define TW    1024
#define THH   1024
#define HW    (TW * THH)
#define NB    8

// -------- CDNA5 async global->LDS helpers (gfx1250) ------------------------
// LDS byte offset = low 32 bits of the generic address of a __shared__ object
// (ISA: LDS aperture truncates addr[31:0]; AS3 offsets live in the low bits).
__device__ __forceinline__ unsigned ldsOff(const void* p) {
    return (unsigned)(uintptr_t)p;
}

__device__ __forceinline__ void asyncLoadB64(void* ldsDst, const void* gsrc) {
    unsigned l = ldsOff(ldsDst);
    uint64_t a = (uint64_t)(uintptr_t)gsrc;
    asm volatile("global_load_async_to_lds_b64 %0, %1, off"
                 :: "v"(l), "v"(a) : "memory");
}
__device__ __forceinline__ void asyncLoadB32(void* ldsDst, const void* gsrc) {
    unsigned l = ldsOff(ldsDst);
    uint64_t a = (uint64_t)(uintptr_t)gsrc;
    asm volatile("global_load_async_to_lds_b32 %0, %1, off"
                 :: "v"(l), "v"(a) : "memory");
}
__device__ __forceinline__ void waitAsync0() {
    asm volatile("s_wait_asynccnt 0" ::: "memory");
}

// gfx1250 has V_TANH_F32 (transcendental unit). 554M tanh evals dominate VALU
// time in this operator; stderr reports which path this toolchain provides.
#if __has_builtin(__builtin_amdgcn_tanhf)
#pragma message("CDNA5: using __builtin_amdgcn_tanhf (v_tanh_f32)")
__device__ __forceinline__ float fast_tanh(float x) {
    return __builtin_amdgcn_tanhf(x);
}
#else
#pragma message("CDNA5: __builtin_amdgcn_tanhf NOT available, ocml tanhf fallback")
__device__ __forceinline__ float fast_tanh(float x) {
    return tanhf(x);
}
#endif

// ===========================================================================
// Kernel 1: Sobel (cross-correlation, zero pad) + tangent init + global max.
// Straight-line: clamped addresses + validity-mask multiplies (no exec
// branches per tap).  Writes tang (interleaved float2), raw mag, atomic max.
// ===========================================================================
__global__ __launch_bounds__(256) void etf_prep(const float* __restrict__ in,
                                                float2* __restrict__ tang,
                                                float* __restrict__ mag,
                                                unsigned* __restrict__ maxBits)
{
    int p = blockIdx.x * 256 + threadIdx.x;       // 0 .. 8M-1
    int b = p >> 20;
    int r = p & (HW - 1);
    int y = r >> 10;
    int x = r & (TW - 1);
    const float* pl = in + (size_t)b * HW;

    int ym = y > 0 ? y - 1 : 0;
    int yp = y < THH - 1 ? y + 1 : THH - 1;
    int xm = x > 0 ? x - 1 : 0;
    int xp = x < TW - 1 ? x + 1 : TW - 1;
    float wyt = (y > 0)       ? 1.0f : 0.0f;   // top row valid
    float wyb = (y < THH - 1) ? 1.0f : 0.0f;   // bottom row valid
    float wxl = (x > 0)       ? 1.0f : 0.0f;   // left col valid
    float wxr = (x < TW - 1)  ? 1.0f : 0.0f;   // right col valid

    const float* rt = pl + (size_t)ym * TW;
    const float* rc = pl + (size_t)y  * TW;
    const float* rb = pl + (size_t)yp * TW;

    float v00 = rt[xm] * (wyt * wxl), v01 = rt[x] * wyt, v02 = rt[xp] * (wyt * wxr);
    float v10 = rc[xm] * wxl,                            v12 = rc[xp] * wxr;
    float v20 = rb[xm] * (wyb * wxl), v21 = rb[x] * wyb, v22 = rb[xp] * (wyb * wxr);

    // XLA conv = cross-correlation (no kernel flip)
    float gx = (v02 - v00) + 2.0f * (v12 - v10) + (v22 - v20);
    float gy = (v20 - v00) + 2.0f * (v21 - v01) + (v22 - v02);

    float m = sqrtf(gx * gx + gy * gy);
    float2 t;
    if (m > 0.0f) { t.x = -gy / m; t.y = gx / m; }  // tang = (-sobel_y, sobel_x)/|.|
    else          { t.x = 0.0f;    t.y = 0.0f;   }

    mag[p]  = m;
    tang[p] = t;

    // block max reduce -> one atomic per block (mag >= 0 so uint-bit compare OK)
    __shared__ float red[256];
    red[threadIdx.x] = m;
    __syncthreads();
#pragma unroll
    for (int s = 128; s > 0; s >>= 1) {
        if (threadIdx.x < s) red[threadIdx.x] = fmaxf(red[threadIdx.x], red[threadIdx.x + s]);
        __syncthreads();
    }
    if (threadIdx.x == 0) atomicMax(maxBits, __float_as_uint(red[0]));
}

// 11-tap accumulation bodies (macros so LDS accesses keep address space 3 and
// stay ds_ ops; CHECKED is a compile-time constant folded by the compiler).
#define V_TAPS(CHECKED)                                                        \
    _Pragma("unroll")                                                          \
    for (int d = -MU; d <= MU; ++d) {                                          \
        if (CHECKED) { int yy = y + d; if (yy < 0 || yy >= THH) continue; }    \
        float2 tY = tS[ctr + d * VTX];                                         \
        float  sY = mS[ctr + d * VTX];                                         \
        float  w  = (fast_tanh((sY - sX) * invM) + 1.0f) *                     \
                    (tX.x * tY.x + tX.y * tY.y) * 0.5f;                        \
        ax += tY.x * w;                                                        \
        ay += tY.y * w;                                                        \
    }

#define H_TAPS(CHECKED)                                                        \
    _Pragma("unroll")                                                          \
    for (int d = -MU; d <= MU; ++d) {                                          \
        if (CHECKED) { int xx = x + d; if (xx < 0 || xx >= TW) continue; }     \
        float2 tY = tS[tid + MU + d];                                          \
        float  sY = mS[tid + MU + d];                                          \
        float  w  = (fast_tanh((sY - sX) * invM) + 1.0f) *                     \
                    (tX.x * tY.x + tX.y * tY.y) * 0.5f;                        \
        ax += tY.x * w;                                                        \
        ay += tY.y * w;                                                        \
    }

// ===========================================================================
// Kernel 2: vertical 11-tap pass.  Tile 32 wide x 16 tall (2 rows/thread),
// halo 5 rows top/bottom -> 26 LDS rows per 16 outputs (1.63x fill) staged
// via async global->LDS.  Block-uniform interior/boundary specialization:
// only 2 of 64 row-tiles pay per-tap boundary VALU.  Writes raw acc.
// ===========================================================================
#define VTX   32
#define VTYO  16                       // output rows per block
#define VROWS (VTYO + 2 * MU)          // 26 staged rows

__global__ __launch_bounds__(256) void etf_pass_v(const float2* __restrict__ tin,
                                                  float2* __restrict__ tout,
                                                  const float* __restrict__ mag,
                                                  const unsigned* __restrict__ maxBits)
{
    __shared__ float2 tS[VROWS * VTX];
    __shared__ float  mS[VROWS * VTX];

    int b  = blockIdx.z;
    int y0 = blockIdx.y * VTYO;
    int x0 = blockIdx.x * VTX;
    size_t base = (size_t)b * HW;
    int tid = threadIdx.y * VTX + threadIdx.x;

    for (int i = tid; i < VROWS * VTX; i += 256) {
        int rr = i >> 5, cc = i & (VTX - 1);
        int gy = y0 - MU + rr;
        gy = gy < 0 ? 0 : (gy > THH - 1 ? THH - 1 : gy);   // clamp address only
        size_t off = base + (size_t)gy * TW + (size_t)(x0 + cc);
        asyncLoadB64(&tS[i], tin + off);
        asyncLoadB32(&mS[i], mag + off);
    }
    waitAsync0();
    __syncthreads();

    int x = x0 + threadIdx.x;
    float invM = ETA / __uint_as_float(*maxBits);
    bool interior = (y0 >= MU) && (y0 + VTYO - 1 + MU < THH);  // block-uniform

#pragma unroll
    for (int k = 0; k < 2; ++k) {
        int ty = threadIdx.y + k * 8;
        int y  = y0 + ty;
        int ctr = (ty + MU) * VTX + threadIdx.x;
        float2 tX = tS[ctr];
        float  sX = mS[ctr];

        float ax = 0.0f, ay = 0.0f;
        if (interior) { V_TAPS(false) }     // no per-tap bounds VALU
        else          { V_TAPS(true)  }     // zero-pad tap => 0 contrib, skip
        float2 o; o.x = ax; o.y = ay;
        tout[base + (size_t)y * TW + x] = o;
    }
}

// ===========================================================================
// Kernel 3: horizontal 11-tap pass (ends an iteration -> fused normalize).
// 256-wide row tile + 5-px halo each side staged via async global->LDS.
// FINAL=true writes planar (B,2,H,W) fp32 to d_out; else interleaved float2.
// ===========================================================================
#define HTX 256

template <bool FINAL>
__global__ __launch_bounds__(256) void etf_pass_h(const float2* __restrict__ tin,
                                                  float2* __restrict__ tout,
                                                  const float* __restrict__ mag,
                                                  const unsigned* __restrict__ maxBits,
                                                  float* __restrict__ outPlanar)
{
    __shared__ float2 tS[HTX + 2 * MU];
    __shared__ float  mS[HTX + 2 * MU];

    int b  = blockIdx.z;
    int y  = blockIdx.y;
    int x0 = blockIdx.x * HTX;
    size_t rowb = (size_t)b * HW + (size_t)y * TW;
    int tid = threadIdx.x;

    for (int i = tid; i < HTX + 2 * MU; i += HTX) {
        int gx = x0 - MU + i;
        gx = gx < 0 ? 0 : (gx > TW - 1 ? TW - 1 : gx);     // clamp address only
        asyncLoadB64(&tS[i], tin + rowb + gx);
        asyncLoadB32(&mS[i], mag + rowb + gx);
    }
    waitAsync0();
    __syncthreads();

    int x = x0 + tid;
    float2 tX = tS[tid + MU];
    float  sX = mS[tid + MU];
    float invM = ETA / __uint_as_float(*maxBits);
    bool interior = (x0 >= MU) && (x0 + HTX - 1 + MU < TW);   // block-uniform

    float ax = 0.0f, ay = 0.0f;
    if (interior) { H_TAPS(false) }
    else          { H_TAPS(true)  }

    // per-pixel renormalization: every iteration ends with the H pass
    float nm = sqrtf(ax * ax + ay * ay);
    if (nm > 0.0f) { ax /= nm; ay /= nm; }

    if (FINAL) {
        size_t pix = (size_t)y * TW + (size_t)x;
        outPlanar[(size_t)b * 2 * HW + pix]      = ax;     // channel 0
        outPlanar[(size_t)b * 2 * HW + HW + pix] = ay;     // channel 1
    } else {
        float2 o; o.x = ax; o.y = ay;
        tout[rowb + x] = o;
    }
}

// ===========================================================================
// Launch: prep -> 3x (V, H).  Ping-pong: d_out region doubles as one tang
// buffer (64 MB, same bytes as the planar output) so d_ws only needs
// mag(32MB) + tangA(64MB) + 4B  ~= 96 MB.  Working set ~160MB ~= L2 (192MB),
// so passes 2..6 are mostly L2-resident -> VALU(tanh)-limited, hence the
// v_tanh_f32 + async-LDS staging + interior-block specialization.
// ===========================================================================
extern "C" void kernel_launch(void* const* d_in, const int* in_sizes, int n_in,
                              void* d_out, int out_size, void* d_ws, size_t ws_size,
                              hipStream_t stream)
{
    const float* in = (const float*)d_in[0];

    char* ws = (char*)d_ws;
    float*    mag     = (float*)ws;                                      // 32 MB
    float2*   tA      = (float2*)(ws + (size_t)NB * HW * sizeof(float)); // 64 MB
    unsigned* maxBits = (unsigned*)(ws + (size_t)NB * HW * sizeof(float)
                                       + (size_t)NB * HW * sizeof(float2));
    float2*   tB      = (float2*)d_out;   // scratch during intermediate passes

    hipMemsetAsync(maxBits, 0, sizeof(unsigned), stream);

    etf_prep<<<(NB * HW) / 256, 256, 0, stream>>>(in, tB, mag, maxBits);

    dim3 vGrid(TW / VTX, THH / VTYO, NB), vBlk(VTX, 8);
    dim3 hGrid(TW / HTX, THH, NB);

    for (int it = 0; it < ITERS; ++it) {
        etf_pass_v<<<vGrid, vBlk, 0, stream>>>(tB, tA, mag, maxBits);
        if (it < ITERS - 1) {
            etf_pass_h<false><<<hGrid, HTX, 0, stream>>>(tA, tB, mag, maxBits, nullptr);
        } else {
            etf_pass_h<true><<<hGrid, HTX, 0, stream>>>(tA, nullptr, mag, maxBits,
                                                        (float*)d_out);
        }
    }
}